// GroupBimodalCSRPool_75265006895913
// MI455X (gfx1250) — compile-verified
//
#include <hip/hip_runtime.h>
#include <hip/hip_bf16.h>

typedef float v2f __attribute__((ext_vector_type(2)));
typedef float v8f __attribute__((ext_vector_type(8)));

#define EPS_SM 1e-12f
#define EPS_BN 1e-5f

// ---------------------------------------------------------------------------
// K1: wave per group. Lanes = 32 projection channels.
//   - segment min / max of x_proj (coalesced)
//   - group constant hc[g][j] = b1[j] + rsqrt(size)*W1[96][j]
//                               - sum_c mn[c]*W1[32+c][j] - sum_c mx[c]*W1[64+c][j]
//   - writes seg[v] = g for all rows of the group
// ---------------------------------------------------------------------------
__global__ void gbp_k1_group(const float* __restrict__ x_proj,
                             const int*   __restrict__ csr,
                             const float* __restrict__ W1,   // [97,4]
                             const float* __restrict__ b1,   // [4]
                             float*       __restrict__ hc,   // [N,4]
                             int*         __restrict__ seg,  // [V]
                             int N)
{
    int wave = (int)((blockIdx.x * blockDim.x + threadIdx.x) >> 5);
    int lane = (int)(threadIdx.x & 31);
    if (wave >= N) return;

    int start = csr[wave];
    int end   = csr[wave + 1];
    int size  = end - start;

    for (int r = lane; r < size; r += 32) seg[start + r] = wave;

    float mn =  __builtin_inff();
    float mx = -__builtin_inff();
    const float* xp = x_proj + (size_t)start * 32 + lane;
    for (int r = 0; r < size; ++r) {
        float v = xp[(size_t)r * 32];
        mn = fminf(mn, v);
        mx = fmaxf(mx, v);
    }

    // per-lane contribution to the 4 dot products
    float t0 = -(mn * W1[(32 + lane) * 4 + 0] + mx * W1[(64 + lane) * 4 + 0]);
    float t1 = -(mn * W1[(32 + lane) * 4 + 1] + mx * W1[(64 + lane) * 4 + 1]);
    float t2 = -(mn * W1[(32 + lane) * 4 + 2] + mx * W1[(64 + lane) * 4 + 2]);
    float t3 = -(mn * W1[(32 + lane) * 4 + 3] + mx * W1[(64 + lane) * 4 + 3]);
    for (int m = 16; m >= 1; m >>= 1) {
        t0 += __shfl_xor(t0, m, 32);
        t1 += __shfl_xor(t1, m, 32);
        t2 += __shfl_xor(t2, m, 32);
        t3 += __shfl_xor(t3, m, 32);
    }
    if (lane == 0) {
        float rsq = rsqrtf((float)size);
        float4 o;
        o.x = t0 + b1[0] + rsq * W1[96 * 4 + 0];
        o.y = t1 + b1[1] + rsq * W1[96 * 4 + 1];
        o.z = t2 + b1[2] + rsq * W1[96 * 4 + 2];
        o.w = t3 + b1[3] + rsq * W1[96 * 4 + 3];
        *(float4*)(hc + (size_t)wave * 4) = o;
    }
}

// ---------------------------------------------------------------------------
// K2: wave per 16-row tile of x_proj. h[V,4] = x_proj @ Wsum + hc[seg] via
// 8 chained V_WMMA_F32_16X16X4_F32 (M=16 rows, K=32 in chunks of 4, N=4 of 16
// columns used). Also accumulates per-block BatchNorm sum / sumsq partials.
//
// No branches around the WMMA inputs: out-of-range A rows are handled by
// clamping the row address (their D rows are simply never stored), and the
// B-fragment zeroing uses value selects with clamped (always in-bounds)
// W1 indices. EXEC is all ones at every WMMA as the ISA requires.
// ---------------------------------------------------------------------------
__global__ void gbp_k2_wmma(const float* __restrict__ x_proj,  // [V,32]
                            const float* __restrict__ W1,      // [97,4]
                            const float* __restrict__ hc,      // [N,4]
                            const int*   __restrict__ seg,     // [V]
                            float*       __restrict__ hbuf,    // [V,4]
                            float*       __restrict__ partials,// [gridDim.x*8]
                            int V)
{
    __shared__ float red[8];
    int tid = (int)threadIdx.x;
    if (tid < 8) red[tid] = 0.0f;
    __syncthreads();

    int lane = tid & 31;
    int tile = (int)(blockIdx.x * (blockDim.x >> 5) + (tid >> 5));
    int base = tile * 16;
    int m    = lane & 15;   // A row within tile / D column index
    int half = lane >> 4;

    bool colActive = (m < 4);
    int  mm        = colActive ? m : 0;   // clamped, always in-bounds for W1

    // B fragments (Wsum chunk): VGPR0 holds K rows {k0, k0+2(hi half)},
    // VGPR1 rows {k0+1, k0+3}; column n = lane&15, only n<4 nonzero.
    v2f bfrag[8];
#pragma unroll
    for (int c = 0; c < 8; ++c) {
        int k0 = 4 * c + 2 * half;
        float bx = W1[(k0    ) * 4 + mm] + W1[(32 + k0    ) * 4 + mm] + W1[(64 + k0    ) * 4 + mm];
        float by = W1[(k0 + 1) * 4 + mm] + W1[(32 + k0 + 1) * 4 + mm] + W1[(64 + k0 + 1) * 4 + mm];
        v2f b;
        b.x = colActive ? bx : 0.0f;
        b.y = colActive ? by : 0.0f;
        bfrag[c] = b;
    }

    // A fragments: lane (both halves) holds row base+m; half selects K pair.
    // Clamp the row: invalid rows produce garbage D rows that are never stored.
    int rowA = base + m;
    if (rowA > V - 1) rowA = V - 1;
    const float* xr = x_proj + (size_t)rowA * 32 + 2 * half;

    v8f acc = {};
#pragma unroll
    for (int c = 0; c < 8; ++c) {
        v2f a = *(const v2f*)(xr + 4 * c);   // 8-byte aligned -> global_load_b64
        acc = __builtin_amdgcn_wmma_f32_16x16x4_f32(
            false, a, false, bfrag[c], (short)0, acc, false, false);
    }

    // D layout: vgpr i, lanes 0-15 -> row i col lane; lanes 16-31 -> row 8+i.
    if (colActive) {
        float s = 0.0f, s2 = 0.0f;
#pragma unroll
        for (int i = 0; i < 8; ++i) {
            int row = base + 8 * half + i;
            if (row < V) {
                int g = seg[row];
                float val = acc[i] + hc[(size_t)g * 4 + m];
                hbuf[(size_t)row * 4 + m] = val;
                s  += val;
                s2 += val * val;
            }
        }
        atomicAdd(&red[m], s);
        atomicAdd(&red[4 + m], s2);
    }
    __syncthreads();
    if (tid < 8) partials[(size_t)blockIdx.x * 8 + tid] = red[tid];
}

// ---------------------------------------------------------------------------
// K2b: single block; reduce partials -> fused BN affine: a[j]=gamma*rsqrt(var+eps),
// b[j]=beta-mean*a  (biased variance, training-mode batch stats).
// ---------------------------------------------------------------------------
__global__ void gbp_k2b_stats(const float* __restrict__ partials, int nblk, int V,
                              const float* __restrict__ gamma,
                              const float* __restrict__ beta,
                              float*       __restrict__ stats) // [8]
{
    __shared__ float sh[256];
    int t  = (int)threadIdx.x;
    int ch = t & 7;
    float s = 0.0f;
    for (int i = t >> 3; i < nblk; i += 32) s += partials[(size_t)i * 8 + ch];
    sh[t] = s;
    __syncthreads();
    for (int step = 128; step >= 8; step >>= 1) {
        if (t < step) sh[t] += sh[t + step];
        __syncthreads();
    }
    if (t < 4) {
        float mean = sh[t] / (float)V;
        float var  = sh[4 + t] / (float)V - mean * mean;
        float a    = gamma[t] * rsqrtf(var + EPS_BN);
        stats[t]     = a;
        stats[4 + t] = beta[t] - mean * a;
    }
}

// ---------------------------------------------------------------------------
// K3: wave per group. Lane owns attention group k = lane>>3 (rep = 128/4 = 32,
// so a float4 at channel 4*lane stays within a single attention group).
// 3 passes over the (<=7) rows: cmax, denom, weighted pooling. Then gating.
// ---------------------------------------------------------------------------
__global__ void gbp_k3_pool(const float* __restrict__ x_mod,  // [V,128]
                            const int*   __restrict__ csr,
                            const float* __restrict__ hbuf,   // [V,4]
                            const float* __restrict__ stats,  // a[4], b[4]
                            const float* __restrict__ Ws,     // [4,4]
                            const float* __restrict__ bs,     // [4]
                            const float* __restrict__ gw,     // [4]
                            const float* __restrict__ gb,     // [4]
                            float*       __restrict__ out,    // [N*128 + N]
                            int N)
{
    int wave = (int)((blockIdx.x * blockDim.x + threadIdx.x) >> 5);
    int lane = (int)(threadIdx.x & 31);
    if (wave >= N) return;

    int start = csr[wave];
    int size  = csr[wave + 1] - start;
    int k     = lane >> 3;

    float a0 = stats[0], a1 = stats[1], a2 = stats[2], a3 = stats[3];
    float c0 = stats[4], c1 = stats[5], c2 = stats[6], c3 = stats[7];
    float w0 = Ws[0 * 4 + k], w1 = Ws[1 * 4 + k], w2 = Ws[2 * 4 + k], w3 = Ws[3 * 4 + k];
    float bsk = bs[k];
    float rsq = rsqrtf((float)size);

    const float4* hp = (const float4*)hbuf + start;

    // pass 1: per-attention-group max score
    float cmax = -__builtin_inff();
    for (int r = 0; r < size; ++r) {
        float4 h = hp[r];
        float h0 = fmaxf(h.x * a0 + c0, 0.0f);
        float h1 = fmaxf(h.y * a1 + c1, 0.0f);
        float h2 = fmaxf(h.z * a2 + c2, 0.0f);
        float h3 = fmaxf(h.w * a3 + c3, 0.0f);
        float C  = h0 * w0 + h1 * w1 + h2 * w2 + h3 * w3 + bsk;
        cmax = fmaxf(cmax, C);
    }

    // pass 2: softmax denominator (group-scaled)
    float denom = EPS_SM;
    for (int r = 0; r < size; ++r) {
        float4 h = hp[r];
        float h0 = fmaxf(h.x * a0 + c0, 0.0f);
        float h1 = fmaxf(h.y * a1 + c1, 0.0f);
        float h2 = fmaxf(h.z * a2 + c2, 0.0f);
        float h3 = fmaxf(h.w * a3 + c3, 0.0f);
        float C  = h0 * w0 + h1 * w1 + h2 * w2 + h3 * w3 + bsk;
        denom += expf((C - cmax) * rsq);
    }
    float inv = 1.0f / denom;

    // pass 3: attention-weighted pooling of x_mod (coalesced float4 per row)
    float ax = 0.0f, ay = 0.0f, az = 0.0f, aw = 0.0f;
    const float4* xm = (const float4*)x_mod + (size_t)start * 32 + lane;
    for (int r = 0; r < size; ++r) {
        float4 h = hp[r];
        float h0 = fmaxf(h.x * a0 + c0, 0.0f);
        float h1 = fmaxf(h.y * a1 + c1, 0.0f);
        float h2 = fmaxf(h.z * a2 + c2, 0.0f);
        float h3 = fmaxf(h.w * a3 + c3, 0.0f);
        float C  = h0 * w0 + h1 * w1 + h2 * w2 + h3 * w3 + bsk;
        float A  = expf((C - cmax) * rsq) * inv;
        float4 x = xm[(size_t)r * 32];
        ax += x.x * A; ay += x.y * A; az += x.z * A; aw += x.w * A;
    }

    // gating on per-group max score
    float g = tanhf(fmaxf(gw[k] * cmax + gb[k], 0.0f));
    float4 o; o.x = ax * g; o.y = ay * g; o.z = az * g; o.w = aw * g;
    ((float4*)out)[(size_t)wave * 32 + lane] = o;
    if (lane == 0) out[(size_t)N * 128 + wave] = (size > 0) ? 1.0f : 0.0f;
}

// ---------------------------------------------------------------------------
extern "C" void kernel_launch(void* const* d_in, const int* in_sizes, int n_in,
                              void* d_out, int out_size, void* d_ws, size_t ws_size,
                              hipStream_t stream) {
    const float* x_mod  = (const float*)d_in[0];
    const float* x_proj = (const float*)d_in[1];
    const int*   csr    = (const int*)d_in[2];   // JAX default: int32
    const float* W1     = (const float*)d_in[3];
    const float* b1     = (const float*)d_in[4];
    const float* gamma  = (const float*)d_in[5];
    const float* beta   = (const float*)d_in[6];
    const float* Ws     = (const float*)d_in[7];
    const float* bs     = (const float*)d_in[8];
    const float* gw     = (const float*)d_in[9];
    const float* gb     = (const float*)d_in[10];

    int V = in_sizes[0] / 128;
    int N = in_sizes[2] - 1;

    int tiles   = (V + 15) / 16;
    int nblk2   = (tiles + 7) / 8;    // K2: 8 waves (tiles) per 256-thread block
    int blocksG = (N + 7) / 8;        // K1/K3: 8 groups per 256-thread block

    // ws layout (16B-aligned pieces; hbuf first so float4 loads are aligned)
    float* hbuf     = (float*)d_ws;                       // V*4 floats
    float* hc       = hbuf + (size_t)V * 4;               // N*4 floats (16B offset)
    int*   seg      = (int*)(hc + (size_t)N * 4);         // V ints
    float* partials = (float*)(seg + V);                  // nblk2*8 floats
    float* stats    = partials + (size_t)nblk2 * 8;       // 8 floats

    gbp_k1_group<<<blocksG, 256, 0, stream>>>(x_proj, csr, W1, b1, hc, seg, N);
    gbp_k2_wmma<<<nblk2, 256, 0, stream>>>(x_proj, W1, hc, seg, hbuf, partials, V);
    gbp_k2b_stats<<<1, 256, 0, stream>>>(partials, nblk2, V, gamma, beta, stats);
    gbp_k3_pool<<<blocksG, 256, 0, stream>>>(x_mod, csr, hbuf, stats, Ws, bs, gw, gb,
                                             (float*)d_out, N);
}